// LearnablePatchAgg_31920196944420
// MI455X (gfx1250) — compile-verified
//
#include <hip/hip_runtime.h>
#include <hip/hip_bf16.h>
#include <stdint.h>
#include <math.h>

typedef float v4f __attribute__((ext_vector_type(4)));

// sizes
static constexpr size_t PVOL = 96ull * 96ull * 96ull;      // 884736, one patch channel
static constexpr size_t OVOL = 192ull * 192ull * 192ull;   // 7077888, one output channel
static constexpr int    GVOL = 48 * 48 * 48;               // 110592, one global_logit channel

// Which of the starts {0,48,96} cover coordinate v (patch extent 96)?
__device__ __forceinline__ void cover(int v, int si[2], int& n) {
    int hi = v / 48; if (hi > 2) hi = 2;
    int lo = hi - 1;
    if (lo >= 0 && (v - lo * 48) < 96) { si[0] = lo; si[1] = hi; n = 2; }
    else                               { si[0] = hi; si[1] = hi; n = 1; }
}

__global__ __launch_bounds__(256)
void patch_agg_kernel(const float* __restrict__ patches,   // (27,4,96,96,96)
                      const float* __restrict__ gl,        // (1,4,48,48,48)
                      const float* __restrict__ cwraw,     // (1,4,1,1,1)
                      const float* __restrict__ fixedwt,   // (1,1,96,96,96)
                      float* __restrict__ out)             // (1,4,192,192,192)
{
    __shared__ __align__(16) float s_w1[96];  // 1D gaussian row (w1[k]/w1max)
    __shared__ float s_cw[8];                 // [c]=sigmoid, [4+c]=1-sigmoid

    const unsigned tid = threadIdx.x;

    // ---- Stage the 1D weight row into LDS with the CDNA5 async global->LDS path.
    // fixed_wt[47][47][k] = w1[k]/w1max, and (w1[i]/m)(w1[j]/m)(w1[k]/m) == fixed_wt[i][j][k].
    if (tid < 96) {
        unsigned ldsoff = (unsigned)(uintptr_t)(&s_w1[tid]);             // low 32 bits == LDS offset
        unsigned long long ga =
            (unsigned long long)(uintptr_t)(fixedwt + ((47 * 96 + 47) * 96) + tid);
        asm volatile("global_load_async_to_lds_b32 %0, %1, off"
                     :: "v"(ldsoff), "v"(ga) : "memory");
    } else if (tid < 100) {
        int c = tid - 96;
        float v = cwraw[c];
        v = fminf(10.0f, fmaxf(-1.0f, v));
        float cw = 1.0f / (1.0f + expf(-v));
        s_cw[c] = cw;
        s_cw[4 + c] = 1.0f - cw;
    }
    asm volatile("s_wait_asynccnt 0" ::: "memory");
    __syncthreads();

    // ---- Thread mapping: x uniform per block; 256 threads * 36 y-blocks sweep (y, z/4).
    const int x  = blockIdx.x;                       // 0..191
    const int t  = blockIdx.y * 256 + tid;           // 0..9215
    const int y  = t / 48;                           // 0..191
    const int zg = t % 48;                           // z group
    const int z0 = zg * 4;

    int sxi[2], syi[2], szi[2];
    int nx, ny, nz;
    cover(x,  sxi, nx);      // uniform per block -> scalar branch
    cover(y,  syi, ny);
    cover(z0, szi, nz);      // whole 4-wide z group shares one coverage set

    float wxv[2], wyv[2];
    v4f   wzv[2];
    #pragma unroll
    for (int i = 0; i < 2; ++i) {
        if (i < nx) wxv[i] = s_w1[x - sxi[i] * 48];
        if (i < ny) wyv[i] = s_w1[y - syi[i] * 48];
        if (i < nz) wzv[i] = *(const v4f*)(s_w1 + (z0 - szi[i] * 48));   // ds_load_b128
    }

    v4f acc0 = {0,0,0,0}, acc1 = {0,0,0,0}, acc2 = {0,0,0,0}, acc3 = {0,0,0,0};
    v4f wm   = {1e-20f, 1e-20f, 1e-20f, 1e-20f};

    // ---- Overlap-add gather: each patch element touched exactly once, streamed NT.
    for (int ix = 0; ix < nx; ++ix) {
        const size_t xoff = (size_t)(x - sxi[ix] * 48) * (96 * 96);
        for (int iy = 0; iy < ny; ++iy) {
            const float  wxy  = wxv[ix] * wyv[iy];
            const size_t xyof = xoff + (size_t)(y - syi[iy] * 48) * 96;
            const int    pxy  = sxi[ix] * 9 + syi[iy] * 3;
            for (int iz = 0; iz < nz; ++iz) {
                const v4f w = wzv[iz] * wxy;
                wm += w;
                const size_t base = (size_t)(pxy + szi[iz]) * (4 * PVOL)
                                  + xyof + (size_t)(z0 - szi[iz] * 48);
                const float* pb = patches + base;
                v4f a0 = __builtin_nontemporal_load((const v4f*)(pb));
                v4f a1 = __builtin_nontemporal_load((const v4f*)(pb + PVOL));
                v4f a2 = __builtin_nontemporal_load((const v4f*)(pb + 2 * PVOL));
                v4f a3 = __builtin_nontemporal_load((const v4f*)(pb + 3 * PVOL));
                acc0 += w * a0; acc1 += w * a1; acc2 += w * a2; acc3 += w * a3;
            }
        }
    }

    const v4f inv = {1.0f / wm.x, 1.0f / wm.y, 1.0f / wm.z, 1.0f / wm.w};

    // ---- Trilinear upsample of global_logit (half-pixel, edge clamp).
    const float fx = (float)x * 0.25f - 0.375f;
    const int   x0 = (int)floorf(fx);
    const float tx = fx - (float)x0;
    const int   x0c = min(47, max(0, x0)), x1c = min(47, max(0, x0 + 1));

    const float fy = (float)y * 0.25f - 0.375f;
    const int   y0 = (int)floorf(fy);
    const float ty = fy - (float)y0;
    const int   y0c = min(47, max(0, y0)), y1c = min(47, max(0, y0 + 1));

    const int zA = max(0, zg - 1), zB = zg, zC = min(47, zg + 1);

    const float w00 = (1.0f - tx) * (1.0f - ty), w01 = (1.0f - tx) * ty;
    const float w10 = tx * (1.0f - ty),          w11 = tx * ty;
    const int r00 = (x0c * 48 + y0c) * 48, r01 = (x0c * 48 + y1c) * 48;
    const int r10 = (x1c * 48 + y0c) * 48, r11 = (x1c * 48 + y1c) * 48;

    v4f upc[4];
    #pragma unroll
    for (int c = 0; c < 4; ++c) {
        const float* g = gl + c * GVOL;
        v4f s = {0,0,0,0};
        const int   rr[4] = {r00, r01, r10, r11};
        const float ww[4] = {w00, w01, w10, w11};
        #pragma unroll
        for (int m = 0; m < 4; ++m) {
            const float* r = g + rr[m];
            const float a = r[zA], b = r[zB], cc = r[zC];
            // z fracs for the 4 outputs of this group: .625,.875 (from zA,zB) .125,.375 (zB,zC)
            v4f v = {0.375f * a + 0.625f * b,
                     0.125f * a + 0.875f * b,
                     0.875f * b + 0.125f * cc,
                     0.625f * b + 0.375f * cc};
            s += ww[m] * v;
        }
        upc[c] = s;
    }

    // ---- Combine: out = cw * (vol/wm) + (1-cw) * upsample, NT streaming stores.
    float* ob = out + (size_t)x * (192 * 192) + (size_t)y * 192 + z0;
    v4f o0 = s_cw[0] * (acc0 * inv) + s_cw[4] * upc[0];
    v4f o1 = s_cw[1] * (acc1 * inv) + s_cw[5] * upc[1];
    v4f o2 = s_cw[2] * (acc2 * inv) + s_cw[6] * upc[2];
    v4f o3 = s_cw[3] * (acc3 * inv) + s_cw[7] * upc[3];
    __builtin_nontemporal_store(o0, (v4f*)(ob));
    __builtin_nontemporal_store(o1, (v4f*)(ob + OVOL));
    __builtin_nontemporal_store(o2, (v4f*)(ob + 2 * OVOL));
    __builtin_nontemporal_store(o3, (v4f*)(ob + 3 * OVOL));
}

extern "C" void kernel_launch(void* const* d_in, const int* in_sizes, int n_in,
                              void* d_out, int out_size, void* d_ws, size_t ws_size,
                              hipStream_t stream) {
    const float* patches = (const float*)d_in[0];   // (27,4,96,96,96) fp32
    const float* gl      = (const float*)d_in[1];   // (1,4,48,48,48)  fp32
    // d_in[2] = starts (27,3) int32 — deterministic {0,48,96} grid, derived arithmetically
    const float* cwraw   = (const float*)d_in[3];   // (1,4,1,1,1)     fp32
    const float* fixedwt = (const float*)d_in[4];   // (1,1,96,96,96)  fp32
    float* out = (float*)d_out;                     // 4*192^3 fp32

    dim3 grid(192, 36);   // x-slab  x  (y,z/4) sweep: 36*256 = 192*48 threads
    dim3 block(256);      // 8 wave32 waves per block
    hipLaunchKernelGGL(patch_agg_kernel, grid, block, 0, stream,
                       patches, gl, cwraw, fixedwt, out);
}